// Spread_37838661878517
// MI455X (gfx1250) — compile-verified
//
#include <hip/hip_runtime.h>

typedef __attribute__((ext_vector_type(2))) float v2f;
typedef __attribute__((ext_vector_type(8))) float v8f;

#define NIN   4096
#define NOUT  4098
#define NBATCH 2048
#define JT    257            // ceil(NOUT/16) j-tiles
#define CPAD  (JT * 16)      // 4112, zero-padded coefficient table length

// ---------------------------------------------------------------------------
// Kernel 1: fold the scan's band of weight + the (1+bias)^p multipliers into a
// 3-tap coefficient table:
//   out[b,j] = x[b,j-2]*C0[j] + x[b,j-1]*C1[j] + x[b,j]*C2[j]
// Term t is introduced at step n=j-2+t and re-multiplied by m=1+bias[j] at
// every later step that touches j, i.e. p = min(j,NIN-1) - n + 1 times.
// Entries with n outside [0,NIN) or j >= NOUT are exactly zero (boundaries).
// ---------------------------------------------------------------------------
__global__ void spread_coef_kernel(const float* __restrict__ weight,
                                   const float* __restrict__ bias,
                                   float* __restrict__ C0,
                                   float* __restrict__ C1,
                                   float* __restrict__ C2) {
    int j = blockIdx.x * blockDim.x + threadIdx.x;
    if (j >= CPAD) return;
    float c0 = 0.f, c1 = 0.f, c2 = 0.f;
    if (j < NOUT) {
        float m = 1.0f + bias[j];
        int last = (j < NIN) ? j : (NIN - 1);
#pragma unroll
        for (int t = 0; t < 3; ++t) {
            int n = j - 2 + t;
            if (n >= 0 && n < NIN) {
                int p = last - n + 1;          // 1..3 multiplications by m
                float mp = m;
                if (p > 1) mp *= m;
                if (p > 2) mp *= m;
                float c = weight[(size_t)j * NIN + n] * mp;
                if (t == 0) c0 = c;
                else if (t == 1) c1 = c;
                else c2 = c;
            }
        }
    }
    C0[j] = c0; C1[j] = c1; C2[j] = c2;
}

// ---------------------------------------------------------------------------
// Kernel 2: one 16(batch) x 16(j) output tile per wave.
// D(16x16) = sum_{s=0..4} A_s(16x4) * B_s(4x16), via v_wmma_f32_16x16x4_f32.
// A_s: x window columns j0-2+4s .. j0+1+4s (clamped; zero coeffs cover edges).
//   32-bit A layout: lanes 0-15 hold M=0..15 with {K=4s,4s+1}; lanes 16-31
//   hold {K=4s+2,4s+3}.
// B_s[k][n] = C_{k-n}[j0+n] when k-n in {0,1,2}, else 0 (branch-free cndmask
//   selects -> EXEC stays all-ones through every WMMA).
// D layout: VGPR r, lane<16 -> row b0+r, lane>=16 -> row b0+8+r, col j0+(lane&15).
// ---------------------------------------------------------------------------
__global__ void spread_wmma_kernel(const float* __restrict__ x,
                                   const float* __restrict__ C0,
                                   const float* __restrict__ C1,
                                   const float* __restrict__ C2,
                                   float* __restrict__ out) {
    int wave = blockIdx.x * (blockDim.x >> 5) + (threadIdx.x >> 5);
    int lane = threadIdx.x & 31;
    int jt = wave % JT;
    int bt = wave / JT;
    int j0 = jt << 4;
    int b0 = bt << 4;
    int half = lane >> 4;       // 0: K even pair / M 0..7 on store
    int l    = lane & 15;       // A: row M=l ; B/D: column N=l

    const float* xrow = x + (size_t)(b0 + l) * NIN;

    // B coefficients for this lane's column (3 loads total, L2-resident table)
    float cc0 = C0[j0 + l];
    float cc1 = C1[j0 + l];
    float cc2 = C2[j0 + l];

    v8f acc = {};
#pragma unroll
    for (int s = 0; s < 5; ++s) {
        int k0 = 4 * s + 2 * half;          // this lane's first K of the pair
        // A: x[b0+l][j0-2+k0], x[b0+l][j0-1+k0]  (clamped; coeffs are 0 there)
        int xc = j0 - 2 + k0;
        xc = xc < 0 ? 0 : xc;
        xc = xc > (NIN - 2) ? (NIN - 2) : xc;
        v2f a;
        a.x = xrow[xc];
        a.y = xrow[xc + 1];
        // B: band matrix entries for rows K=k0, k0+1 at column N=l
        int t0 = k0 - l;
        int t1 = t0 + 1;
        v2f b;
        b.x = (t0 == 0) ? cc0 : (t0 == 1) ? cc1 : (t0 == 2) ? cc2 : 0.f;
        b.y = (t1 == 0) ? cc0 : (t1 == 1) ? cc1 : (t1 == 2) ? cc2 : 0.f;
        acc = __builtin_amdgcn_wmma_f32_16x16x4_f32(
            /*neg_a=*/false, a, /*neg_b=*/false, b,
            /*c_mod=*/(short)0, acc, /*reuse_a=*/false, /*reuse_b=*/false);
    }

    // Store D; column guard only affects the final (partial) j-tile.
    int j = j0 + l;
    if (j < NOUT) {
        float* ocol = out + (size_t)(b0 + 8 * half) * NOUT + j;
#pragma unroll
        for (int r = 0; r < 8; ++r) {
            ocol[(size_t)r * NOUT] = acc[r];
        }
    }
}

extern "C" void kernel_launch(void* const* d_in, const int* in_sizes, int n_in,
                              void* d_out, int out_size, void* d_ws, size_t ws_size,
                              hipStream_t stream) {
    const float* x      = (const float*)d_in[0];   // (2048, 4096)
    const float* weight = (const float*)d_in[1];   // (4098, 4096)
    const float* bias   = (const float*)d_in[2];   // (4098,)
    float* out = (float*)d_out;                    // (2048, 4098)

    float* C0 = (float*)d_ws;                      // 3 * 4112 floats = ~48 KB
    float* C1 = C0 + CPAD;
    float* C2 = C1 + CPAD;

    spread_coef_kernel<<<(CPAD + 255) / 256, 256, 0, stream>>>(weight, bias, C0, C1, C2);

    const int tiles  = (NBATCH / 16) * JT;         // 128 * 257 = 32896
    const int blocks = tiles / 8;                  // 8 waves (tiles) per 256-thr block
    spread_wmma_kernel<<<blocks, 256, 0, stream>>>(x, C0, C1, C2, out);
}